// HeteroGNN_48189533061506
// MI455X (gfx1250) — compile-verified
//
#include <hip/hip_runtime.h>
#include <hip/hip_bf16.h>

typedef __attribute__((ext_vector_type(2))) float v2f;
typedef __attribute__((ext_vector_type(8))) float v8f;

#define NP 200000
#define NA 100000
#define HD 128

// ---------------------------------------------------------------------------
// Generic fused SAGE GEMM:  out = scale * ( sum_t A_t @ W_t^T  + b0 (+ b1) )
// Uses V_WMMA_F32_16X16X4_F32 (fp32 matrix pipe, exact fp32 math).
// Block = 128 threads = 4 waves; each wave owns 2 column tiles (16 cols each),
// the block owns one 16-row stripe.  Row-local => in-place (out == A_t) safe.
// ---------------------------------------------------------------------------
struct GemmCfg {
  const float* A0; const float* W0;
  const float* A1; const float* W1;
  const float* A2; const float* W2;
  const float* A3; const float* W3;
  const float* b0; const float* b1;
  float scale;
  float* out;
};

__global__ __launch_bounds__(128) void sage_gemm_wmma(GemmCfg c) {
  const int lane = threadIdx.x & 31;
  const int wv   = threadIdx.x >> 5;
  const int r    = lane & 15;   // row-in-tile for A, col for B/C
  const int hs   = lane >> 4;   // K-half select (0/1) per ISA f32 frag layout
  const int m0   = blockIdx.x * 16;
  const int h0   = wv * 32;     // this wave's two column tiles: h0, h0+16

  float bs0 = c.b0[h0 + r];
  float bs1 = c.b0[h0 + 16 + r];
  if (c.b1) { bs0 += c.b1[h0 + r]; bs1 += c.b1[h0 + 16 + r]; }

  v8f acc0, acc1;
#pragma unroll
  for (int i = 0; i < 8; ++i) { acc0[i] = bs0; acc1[i] = bs1; }

  const float* As[4] = {c.A0, c.A1, c.A2, c.A3};
  const float* Ws[4] = {c.W0, c.W1, c.W2, c.W3};

#pragma unroll
  for (int t = 0; t < 4; ++t) {
    if (As[t] == nullptr) continue;
    const float* Ar  = As[t] + (size_t)(m0 + r) * HD + 2 * hs;
    const float* Wr0 = Ws[t] + (size_t)(h0 + r) * HD + 2 * hs;
    const float* Wr1 = Ws[t] + (size_t)(h0 + 16 + r) * HD + 2 * hs;
#pragma unroll
    for (int k = 0; k < HD; k += 4) {
      v2f a  = *(const v2f*)(Ar  + k);
      v2f b0 = *(const v2f*)(Wr0 + k);
      v2f b1 = *(const v2f*)(Wr1 + k);
      acc0 = __builtin_amdgcn_wmma_f32_16x16x4_f32(false, a, false, b0,
                                                   (short)0, acc0, false, false);
      acc1 = __builtin_amdgcn_wmma_f32_16x16x4_f32(false, a, false, b1,
                                                   (short)0, acc1, false, false);
    }
  }

#pragma unroll
  for (int v = 0; v < 8; ++v) {
    int row = m0 + v + 8 * hs;
    c.out[(size_t)row * HD + h0 + r]      = acc0[v] * c.scale;
    c.out[(size_t)row * HD + h0 + 16 + r] = acc1[v] * c.scale;
  }
}

// ---------------------------------------------------------------------------
// Edge scatter-add: sums[dst] += h[src]; cnt[dst] += 1.  2 edges / 256-thr blk.
// Relaxed agent-scope f32 atomics -> global_atomic_add_f32 at L2.
// ---------------------------------------------------------------------------
__global__ __launch_bounds__(256) void scatter_add_edges(
    const float* __restrict__ hsrc, const int* __restrict__ src,
    const int* __restrict__ dst, float* __restrict__ sums,
    float* __restrict__ cnt, int E) {
  int e = blockIdx.x * 2 + (threadIdx.x >> 7);
  int t = threadIdx.x & 127;
  if (e >= E) return;
  int s = src[e], d = dst[e];
  float v = hsrc[(size_t)s * HD + t];
  __hip_atomic_fetch_add(&sums[(size_t)d * HD + t], v,
                         __ATOMIC_RELAXED, __HIP_MEMORY_SCOPE_AGENT);
  if (t == 0)
    __hip_atomic_fetch_add(&cnt[d], 1.0f,
                           __ATOMIC_RELAXED, __HIP_MEMORY_SCOPE_AGENT);
}

// sums[row, :] /= max(cnt[row], 1)   (float4 vectorized, i indexes n*32 quads)
__global__ __launch_bounds__(256) void normalize_mean(
    float* __restrict__ sums, const float* __restrict__ cnt, int n) {
  int i = blockIdx.x * blockDim.x + threadIdx.x;
  if (i >= n * 32) return;
  int row = i >> 5;
  float inv = 1.0f / fmaxf(cnt[row], 1.0f);
  float4* p = (float4*)sums;
  float4 v = p[i];
  v.x *= inv; v.y *= inv; v.z *= inv; v.w *= inv;
  p[i] = v;
}

// ---------------------------------------------------------------------------
extern "C" void kernel_launch(void* const* d_in, const int* in_sizes, int n_in,
                              void* d_out, int out_size, void* d_ws, size_t ws_size,
                              hipStream_t stream) {
  (void)out_size; (void)ws_size;
  const float* x_paper  = (const float*)d_in[0];
  const float* x_author = (const float*)d_in[1];
  const int* cs  = (const int*)d_in[2];
  const int* cd  = (const int*)d_in[3];
  const int* wss = (const int*)d_in[4];
  const int* wsd = (const int*)d_in[5];
  const int* wns = (const int*)d_in[6];
  const int* wnd = (const int*)d_in[7];
  const int Ec = in_sizes[2], Ew = in_sizes[4], Ewr = in_sizes[6];

  // ---- parameter leaf mapping (3 possible harness conventions) ----
  const float *enc_pw, *enc_pb, *enc_aw, *enc_ab;
  const float *WL[2][3], *BL[2][3], *WR[2][3];  // et: 0=cites 1=writes 2=written
  const int P = 8;
  if (n_in <= 9) {
    // single concatenated blob, dict insertion order
    const float* q = (const float*)d_in[P];
    enc_pw = q;           enc_pb = q + 16384;
    enc_aw = q + 16512;   enc_ab = q + 32896;
    const float* base = q + 33024;
    for (int L = 0; L < 2; ++L)
      for (int e = 0; e < 3; ++e) {
        const float* b = base + (size_t)L * 98688 + (size_t)e * 32896;
        WL[L][e] = b; BL[L][e] = b + 16384; WR[L][e] = b + 16512;
      }
  } else if (in_sizes[P] == HD * HD) {
    // per-leaf, dict insertion order: w,b pairs then (wl,bl,wr) triples
    enc_pw = (const float*)d_in[P + 0]; enc_pb = (const float*)d_in[P + 1];
    enc_aw = (const float*)d_in[P + 2]; enc_ab = (const float*)d_in[P + 3];
    for (int L = 0; L < 2; ++L)
      for (int e = 0; e < 3; ++e) {
        int b = P + 4 + L * 9 + e * 3;
        WL[L][e] = (const float*)d_in[b];
        BL[L][e] = (const float*)d_in[b + 1];
        WR[L][e] = (const float*)d_in[b + 2];
      }
  } else {
    // per-leaf, alphabetically sorted (jax tree_flatten): b,w then (bl,wl,wr)
    enc_ab = (const float*)d_in[P + 0]; enc_aw = (const float*)d_in[P + 1];
    enc_pb = (const float*)d_in[P + 2]; enc_pw = (const float*)d_in[P + 3];
    for (int L = 0; L < 2; ++L)
      for (int e = 0; e < 3; ++e) {
        int b = P + 4 + L * 9 + e * 3;
        BL[L][e] = (const float*)d_in[b];
        WL[L][e] = (const float*)d_in[b + 1];
        WR[L][e] = (const float*)d_in[b + 2];
      }
  }

  // ---- workspace layout (~257 MB) ----
  char* w = (char*)d_ws;
  float* h_p   = (float*)w; w += (size_t)NP * HD * 4;   // paper features
  float* h_a   = (float*)w; w += (size_t)NA * HD * 4;   // author features
  float* agg_w = (float*)w; w += (size_t)NP * HD * 4;   // writes aggregate
  float* cnt   = (float*)w; w += (size_t)NP * 4;        // shared count buffer

  float* out_p  = (float*)d_out;
  float* out_a  = out_p + (size_t)NP * HD;
  float* agg_c  = out_p;   // d_out paper region doubles as cites aggregate
  float* agg_wr = out_a;   // d_out author region doubles as written aggregate

  // ---- encoder ----
  {
    GemmCfg cfg{ x_paper, enc_pw, nullptr,nullptr, nullptr,nullptr, nullptr,nullptr,
                 enc_pb, nullptr, 1.0f, h_p };
    sage_gemm_wmma<<<NP / 16, 128, 0, stream>>>(cfg);
  }
  {
    GemmCfg cfg{ x_author, enc_aw, nullptr,nullptr, nullptr,nullptr, nullptr,nullptr,
                 enc_ab, nullptr, 1.0f, h_a };
    sage_gemm_wmma<<<NA / 16, 128, 0, stream>>>(cfg);
  }

  for (int L = 0; L < 2; ++L) {
    // cites: paper -> paper
    hipMemsetAsync(agg_c, 0, (size_t)NP * HD * 4, stream);
    hipMemsetAsync(cnt,   0, (size_t)NP * 4, stream);
    scatter_add_edges<<<(Ec + 1) / 2, 256, 0, stream>>>(h_p, cs, cd, agg_c, cnt, Ec);
    normalize_mean<<<(NP * 32 + 255) / 256, 256, 0, stream>>>(agg_c, cnt, NP);
    // writes: author -> paper
    hipMemsetAsync(agg_w, 0, (size_t)NP * HD * 4, stream);
    hipMemsetAsync(cnt,   0, (size_t)NP * 4, stream);
    scatter_add_edges<<<(Ew + 1) / 2, 256, 0, stream>>>(h_a, wss, wsd, agg_w, cnt, Ew);
    normalize_mean<<<(NP * 32 + 255) / 256, 256, 0, stream>>>(agg_w, cnt, NP);
    // written: paper -> author
    hipMemsetAsync(agg_wr, 0, (size_t)NA * HD * 4, stream);
    hipMemsetAsync(cnt,    0, (size_t)NA * 4, stream);
    scatter_add_edges<<<(Ewr + 1) / 2, 256, 0, stream>>>(h_p, wns, wnd, agg_wr, cnt, Ewr);
    normalize_mean<<<(NA * 32 + 255) / 256, 256, 0, stream>>>(agg_wr, cnt, NA);

    float* op = (L == 0) ? h_p : out_p;   // layer0 in-place, layer1 -> d_out
    float* oa = (L == 0) ? h_a : out_a;
    // h_p_new = 0.5*(agg_c@wl_c^T + bl_c + h_p@wr_c^T + agg_w@wl_w^T + bl_w + h_p@wr_w^T)
    GemmCfg cp{ agg_c, WL[L][0], h_p, WR[L][0], agg_w, WL[L][1], h_p, WR[L][1],
                BL[L][0], BL[L][1], 0.5f, op };
    sage_gemm_wmma<<<NP / 16, 128, 0, stream>>>(cp);
    // h_a_new = agg_wr@wl^T + bl + h_a@wr^T
    GemmCfg ca{ agg_wr, WL[L][2], h_a, WR[L][2], nullptr,nullptr, nullptr,nullptr,
                BL[L][2], nullptr, 1.0f, oa };
    sage_gemm_wmma<<<NA / 16, 128, 0, stream>>>(ca);
  }
}